// SimpleOrdinalLinearDecayEmbedding_43198781063507
// MI455X (gfx1250) — compile-verified
//
#include <hip/hip_runtime.h>
#include <hip/hip_bf16.h>

typedef __attribute__((ext_vector_type(2))) float v2f;
typedef __attribute__((ext_vector_type(8))) float v8f;

#define QDIM 800   // n_questions (K of the GEMM)
#define DDIM 64    // embed dim   (N of the GEMM)
#define KCATS 4

// ---------------------------------------------------------------------------
// Kernel 1: per-(b,s) scalar gate = sigmoid(Wr . ordinal_weights(r) + br)
// ---------------------------------------------------------------------------
__global__ void gate_kernel(const int* __restrict__ r_data,
                            const float* __restrict__ Wr,
                            const float* __restrict__ br,
                            float* __restrict__ gate, int M) {
  int i = blockIdx.x * blockDim.x + threadIdx.x;
  if (i >= M) return;
  float r = (float)r_data[i];
  float acc = br[0];
#pragma unroll
  for (int k = 0; k < KCATS; ++k) {
    float w = fmaxf(1.0f - fabsf((float)k - r) * (1.0f / (float)(KCATS - 1)), 0.0f);
    acc = fmaf(Wr[k], w, acc);
  }
  gate[i] = 1.0f / (1.0f + expf(-acc));
}

// ---------------------------------------------------------------------------
// Kernel 2: out[m, d] = gate[m] * (q[m,:] . Wq[d,:]) + bq[d]
//   M = B*S rows, N = DDIM = 64 cols, K = QDIM = 800
//   One wave -> 16 x 64 output tile via V_WMMA_F32_16X16X4_F32.
// ---------------------------------------------------------------------------
#define WMMA_F32X4(ACC, A, B)                                                  \
  ACC = __builtin_amdgcn_wmma_f32_16x16x4_f32(false, (A), false, (B),          \
                                              (short)0, (ACC), false, false)

__launch_bounds__(256)
__global__ void gemm_gate_bias_kernel(const float* __restrict__ q,
                                      const float* __restrict__ Wq,
                                      const float* __restrict__ bq,
                                      const float* __restrict__ gate,
                                      float* __restrict__ out, int M) {
  const int wave = threadIdx.x >> 5;
  const int lane = threadIdx.x & 31;
  const int tile = blockIdx.x * (blockDim.x >> 5) + wave;
  const int M0 = tile * 16;
  if (M0 >= M) return;  // wave-uniform: EXEC stays all-ones for WMMA

  const int lr = lane & 15;     // row-in-tile (A) / col (B,C,D)
  const int half = lane >> 4;   // K-slot half select

  // A: q rows M0+lr, stepping K.  B: Wq rows (= output cols) lr + 16*j.
  const float* aPtr = q + (size_t)(M0 + lr) * QDIM + 4 * half;
  const float* bPtr = Wq + (size_t)lr * QDIM + 4 * half;

  v8f acc0 = {}, acc1 = {}, acc2 = {}, acc3 = {};

  for (int k0 = 0; k0 < QDIM; k0 += 8) {
    float4 a4 = *(const float4*)aPtr;
    float4 b0 = *(const float4*)(bPtr + 0 * 16 * QDIM);
    float4 b1 = *(const float4*)(bPtr + 1 * 16 * QDIM);
    float4 b2 = *(const float4*)(bPtr + 2 * 16 * QDIM);
    float4 b3 = *(const float4*)(bPtr + 3 * 16 * QDIM);
    aPtr += 8;
    bPtr += 8;

    // Slot-permuted but A/B-consistent K coverage: exact fp32 GEMM.
    v2f aLo = {a4.x, a4.y}, aHi = {a4.z, a4.w};
    v2f b0lo = {b0.x, b0.y}, b0hi = {b0.z, b0.w};
    v2f b1lo = {b1.x, b1.y}, b1hi = {b1.z, b1.w};
    v2f b2lo = {b2.x, b2.y}, b2hi = {b2.z, b2.w};
    v2f b3lo = {b3.x, b3.y}, b3hi = {b3.z, b3.w};

    WMMA_F32X4(acc0, aLo, b0lo);
    WMMA_F32X4(acc1, aLo, b1lo);
    WMMA_F32X4(acc2, aLo, b2lo);
    WMMA_F32X4(acc3, aLo, b3lo);
    WMMA_F32X4(acc0, aHi, b0hi);
    WMMA_F32X4(acc1, aHi, b1hi);
    WMMA_F32X4(acc2, aHi, b2hi);
    WMMA_F32X4(acc3, aHi, b3hi);
  }

  // Epilogue: out = gate[row]*acc + bq[col].
  const float bq0 = bq[lr + 0];
  const float bq1 = bq[lr + 16];
  const float bq2 = bq[lr + 32];
  const float bq3 = bq[lr + 48];

#pragma unroll
  for (int r = 0; r < 8; ++r) {
    const int row = M0 + r + 8 * half;       // C/D layout: vgpr r, lane half
    const float g = gate[row];               // broadcast per half-wave
    float* o = out + (size_t)row * DDIM + lr;
    o[0]  = fmaf(acc0[r], g, bq0);
    o[16] = fmaf(acc1[r], g, bq1);
    o[32] = fmaf(acc2[r], g, bq2);
    o[48] = fmaf(acc3[r], g, bq3);
  }
}

// ---------------------------------------------------------------------------
// Launch
// ---------------------------------------------------------------------------
extern "C" void kernel_launch(void* const* d_in, const int* in_sizes, int n_in,
                              void* d_out, int out_size, void* d_ws, size_t ws_size,
                              hipStream_t stream) {
  const float* q_data = (const float*)d_in[0];
  const int*   r_data = (const int*)d_in[1];
  const float* Wq     = (const float*)d_in[2];
  const float* bq     = (const float*)d_in[3];
  const float* Wr     = (const float*)d_in[4];
  const float* br     = (const float*)d_in[5];
  float* out = (float*)d_out;

  const int M = in_sizes[1];        // B*S = 32768 rows
  float* gate = (float*)d_ws;       // M floats of scratch

  // 1) gates
  gate_kernel<<<(M + 255) / 256, 256, 0, stream>>>(r_data, Wr, br, gate, M);

  // 2) WMMA GEMM + gated-bias epilogue. 8 waves/block, one 16x64 tile each.
  const int tiles = (M + 15) / 16;
  const int wavesPerBlock = 8;
  const int blocks = (tiles + wavesPerBlock - 1) / wavesPerBlock;
  gemm_gate_bias_kernel<<<blocks, wavesPerBlock * 32, 0, stream>>>(
      q_data, Wq, bq, gate, out, M);
}